// NUFFTLayerMultiChannelInitOneSided_59906203845045
// MI455X (gfx1250) — compile-verified
//
#include <hip/hip_runtime.h>
#include <math.h>

#define MM    8193      // mesh size (odd)
#define MPAD  8196      // padded to multiple of 4 for K-loop
#define HALFM 4096      // MM // 2
#define NPTS  512
#define NB    8

static constexpr double PI64 = 3.14159265358979323846;
#define TAUF  ((float)(12.0/(8193.0*8193.0)))     // tau = 12*(L/(2*pi*M))^2, L=2*pi
#define LFF   ((float)(2.0*PI64))

typedef __attribute__((ext_vector_type(2))) float v2f;
typedef __attribute__((ext_vector_type(8))) float v8f;

// ---------------------------------------------------------------------------
// CDNA5 async global->LDS copies (ASYNCcnt path).  Generic pointers to
// __shared__ hold the LDS byte offset in their low 32 bits (ISA 10.2).
// ---------------------------------------------------------------------------
__device__ __forceinline__ void async_g2l_b64(void* lds_dst, const void* gsrc) {
  unsigned lds = (unsigned)(unsigned long long)(uintptr_t)lds_dst;
  asm volatile("global_load_async_to_lds_b64 %0, %1, off"
               :: "v"(lds), "v"(gsrc) : "memory");
}
__device__ __forceinline__ void async_g2l_b32(void* lds_dst, const void* gsrc) {
  unsigned lds = (unsigned)(unsigned long long)(uintptr_t)lds_dst;
  asm volatile("global_load_async_to_lds_b32 %0, %1, off"
               :: "v"(lds), "v"(gsrc) : "memory");
}
__device__ __forceinline__ void async_wait0() {
  asm volatile("s_wait_asynccnt 0x0" ::: "memory");
}

// ---------------------------------------------------------------------------
// Kernel 0: one-time tables in workspace.
//   trigG[t] = {cos, sin}(theta*t), t = 0..4096   (theta = 2*pi/M)
//   xgG[p]   = XGRID[p] = -pi + p*2*pi/M
// ---------------------------------------------------------------------------
__global__ __launch_bounds__(256) void k0_tables(float* __restrict__ trigG,
                                                 float* __restrict__ xgG) {
  int t = blockIdx.x * 256 + threadIdx.x;
  double a = (double)t * (2.0 * PI64 / 8193.0);
  if (t <= HALFM) {
    trigG[2 * t + 0] = (float)cos(a);
    trigG[2 * t + 1] = (float)sin(a);
  }
  if (t < MM) xgG[t] = (float)(-PI64 + a);
}

// ---------------------------------------------------------------------------
// Kernel 1: spread points onto mesh.  arr16 is 16 x MPAD row-major:
//   rows 0..7  = arr[b,m] = sum_n normalPer(x[b,n]-xg[m])
//   rows 8..15 = 0, cols >= MM = 0   (zero padding for the WMMA GEMM)
// ---------------------------------------------------------------------------
__global__ __launch_bounds__(256) void k1_spread(const float* __restrict__ x,
                                                 const float* __restrict__ sigmaVar,
                                                 const float* __restrict__ xgG,
                                                 float* __restrict__ arr16) {
  __shared__ float sx[NPTS];
  int row = blockIdx.y;
  if (row < NB) {
    for (int i = threadIdx.x; i < NPTS; i += 256) sx[i] = x[row * NPTS + i];
  }
  __syncthreads();
  int m = blockIdx.x * 256 + threadIdx.x;
  if (m >= MPAD) return;
  float val = 0.0f;
  if (row < NB && m < MM) {
    float xgm = xgG[m];
    float s   = sigmaVar[0];
    float s2  = s * s;
    float inv = 1.0f / (sqrtf(2.0f * (float)PI64) * s);
    float acc = 0.0f;
    for (int j = 0; j < NPTS; ++j) {
      float d  = sx[j] - xgm;
      float dm = d - LFF;
      float dp = d + LFF;
      acc += expf((-0.5f * (d  * d )) / s2)
           + expf((-0.5f * (dm * dm)) / s2)
           + expf((-0.5f * (dp * dp)) / s2);
    }
    val = inv * acc;
  }
  arr16[row * MPAD + m] = val;
}

// ---------------------------------------------------------------------------
// WMMA trig GEMM:  Dc[r,p] = sum_j Ac[r,j] * cos(theta*j*p)
//                  Ds[r,p] = sum_j As[r,j] * sin(theta*j*p)
// A matrices are 16 x MPAD row-major (zero-padded).  One wave32 per 16-col
// p-tile; B tiles generated from a 4097-entry cos/sin LDS table (async-
// staged) via running indices (j*p mod M).  V_WMMA_F32_16X16X4_F32.
// ---------------------------------------------------------------------------
__global__ __launch_bounds__(256) void trig_gemm(const float* Ac, const float* As,
                                                 const float* trigG,
                                                 float* Dc, float* Ds) {
  __shared__ v2f trig[HALFM + 1];      // 4097 * 8B = 32776 B
  for (int t = threadIdx.x; t <= HALFM; t += 256)
    async_g2l_b64(&trig[t], trigG + 2 * t);        // ASYNCcnt global->LDS
  async_wait0();
  __syncthreads();

  int wv   = threadIdx.x >> 5;          // wave id 0..7 (wave32)
  int lane = threadIdx.x & 31;
  int tile = blockIdx.x * 8 + wv;       // 16-column output tile
  if (tile * 16 >= MM) return;          // wave-uniform exit: EXEC stays all-1s

  int n    = lane & 15;                 // A row r == B/D column n
  int hi   = lane >> 4;                 // 0: K=0,1   1: K=2,3
  int koff = hi * 2;
  int p    = tile * 16 + n;
  int pm   = (p < MM) ? p : 0;          // clamp for last partial tile (store masked)

  int step = (int)(((long long)4 * pm) % MM);         // index advance per K-step
  int idx0 = (int)(((long long)koff * pm) % MM);      // (koff  )*p mod M
  int idx1 = idx0 + pm; if (idx1 >= MM) idx1 -= MM;   // (koff+1)*p mod M

  const float* acp = Ac + n * MPAD + koff;
  const float* asp = As + n * MPAD + koff;

  v8f accC = {};
  v8f accS = {};
  for (int j = 0; j < MPAD; j += 4) {
    v2f a_c = *(const v2f*)(acp + j);   // A[r, j+koff], A[r, j+koff+1]
    v2f a_s = *(const v2f*)(asp + j);
    // Branchless symmetry fold: cos(th(M-t))=cos(th t), sin(th(M-t))=-sin(th t)
    bool f0 = idx0 > HALFM;
    bool f1 = idx1 > HALFM;
    int  t0 = f0 ? (MM - idx0) : idx0;
    int  t1 = f1 ? (MM - idx1) : idx1;
    v2f v0 = trig[t0];
    v2f v1 = trig[t1];
    v2f bc; bc.x = v0.x;                 bc.y = v1.x;
    v2f bs; bs.x = f0 ? -v0.y : v0.y;    bs.y = f1 ? -v1.y : v1.y;
    accC = __builtin_amdgcn_wmma_f32_16x16x4_f32(false, a_c, false, bc,
                                                 (short)0, accC, false, false);
    accS = __builtin_amdgcn_wmma_f32_16x16x4_f32(false, a_s, false, bs,
                                                 (short)0, accS, false, false);
    idx0 += step; if (idx0 >= MM) idx0 -= MM;
    idx1 += step; if (idx1 >= MM) idx1 -= MM;
    __builtin_prefetch(acp + j + 256, 0, 1);      // global_prefetch_b8
  }

  if (p < MM) {
    // C/D layout: VGPR g holds row (g + 8*hi), column n = lane&15
    for (int g = 0; g < 8; ++g) {
      int row = g + 8 * hi;
      Dc[row * MPAD + p] = accC[g];
      Ds[row * MPAD + p] = accS[g];
    }
  }
}

// ---------------------------------------------------------------------------
// Kernel 3: spectral pointwise.  G = Ccos+Csin (= ReF-ImF), H = Ccos-Csin.
// Channel swap from ifftshift over the channel axis, index shift (m+4096)%M
// from fftshift/ifftshift over the mesh axis, analytic deconv.
//   Tm[c*8+b, m] =  deconv'[m]*multRe_{1-c}'[m] * G[b,m]
//   mU[c*8+b, m] = -deconv'[m]*multIm_{1-c}'[m] * H[b,m]
// ---------------------------------------------------------------------------
__global__ __launch_bounds__(256) void k3_pointwise(
    const float* __restrict__ Ccos, const float* __restrict__ Csin,
    const float* __restrict__ mRe0, const float* __restrict__ mIm0,
    const float* __restrict__ mRe1, const float* __restrict__ mIm1,
    float* __restrict__ Tm, float* __restrict__ mU) {
  int idx = blockIdx.x * 256 + threadIdx.x;
  if (idx >= NB * MPAD) return;
  int b = idx / MPAD;
  int m = idx - b * MPAD;
  if (m >= MM) {            // zero padding columns
    Tm[b * MPAD + m] = 0.0f;       mU[b * MPAD + m] = 0.0f;
    Tm[(8 + b) * MPAD + m] = 0.0f; mU[(8 + b) * MPAD + m] = 0.0f;
    return;
  }
  float Cc = Ccos[b * MPAD + m], Cs = Csin[b * MPAD + m];
  float G = Cc + Cs, H = Cc - Cs;
  int t = m + HALFM; if (t >= MM) t -= MM;          // shifted multiplier index
  float kv  = (m <= HALFM) ? (float)m : (float)(m - MM);  // kg[(m+4096)%M]
  float dec = sqrtf((float)PI64 / TAUF) * expf((kv * kv) * TAUF);
  float P0 = dec * mRe1[t], Q0 = dec * mIm1[t];     // output channel 0 <- mult 1
  float P1 = dec * mRe0[t], Q1 = dec * mIm0[t];     // output channel 1 <- mult 0
  Tm[b * MPAD + m]       = P0 * G;   mU[b * MPAD + m]       = -(Q0 * H);
  Tm[(8 + b) * MPAD + m] = P1 * G;   mU[(8 + b) * MPAD + m] = -(Q1 * H);
}

// ---------------------------------------------------------------------------
// Kernel 5: interpolate back to points.
//   fmm[b,n,c] = (1/M) * (1/(2*pi*M/L)) * sum_p (Ec+Es)[c*8+b, p] * gaussianPer
// ---------------------------------------------------------------------------
__global__ __launch_bounds__(256) void k5_interp(const float* __restrict__ x,
                                                 const float* __restrict__ xgG,
                                                 const float* __restrict__ Ec,
                                                 const float* __restrict__ Es,
                                                 float* __restrict__ out) {
  __shared__ float xg[MM];
  for (int t = threadIdx.x; t < MM; t += 256)
    async_g2l_b32(&xg[t], xgG + t);                // ASYNCcnt global->LDS
  async_wait0();
  __syncthreads();
  int tid = blockIdx.x * 256 + threadIdx.x;
  if (tid >= NB * NPTS) return;
  int b = tid >> 9;
  int n = tid & (NPTS - 1);
  float xv = x[b * NPTS + n];
  const float* e0c = Ec + b * MPAD;
  const float* e0s = Es + b * MPAD;
  const float* e1c = Ec + (8 + b) * MPAD;
  const float* e1s = Es + (8 + b) * MPAD;
  const float r4t = 1.0f / (4.0f * TAUF);
  float acc0 = 0.0f, acc1 = 0.0f;
  for (int p = 0; p < MM; ++p) {
    float d  = xv - xg[p];
    float dm = d - LFF, dp = d + LFF;
    float g3 = expf(-(d  * d ) * r4t)
             + expf(-(dm * dm) * r4t)
             + expf(-(dp * dp) * r4t);
    acc0 += (e0c[p] + e0s[p]) * g3;
    acc1 += (e1c[p] + e1s[p]) * g3;
  }
  const float invM  = 1.0f / 8193.0f;                       // ifft 1/M
  const float denom = (float)(2.0 * PI64 * 8193.0) / LFF;   // 2*pi*M/L
  out[(b * NPTS + n) * 2 + 0] = (acc0 * invM) / denom;
  out[(b * NPTS + n) * 2 + 1] = (acc1 * invM) / denom;
}

// ---------------------------------------------------------------------------
extern "C" void kernel_launch(void* const* d_in, const int* in_sizes, int n_in,
                              void* d_out, int out_size, void* d_ws, size_t ws_size,
                              hipStream_t stream) {
  (void)in_sizes; (void)n_in; (void)out_size; (void)ws_size;
  const float* x    = (const float*)d_in[0];
  const float* sig  = (const float*)d_in[1];
  const float* mRe0 = (const float*)d_in[2];
  const float* mIm0 = (const float*)d_in[3];
  const float* mRe1 = (const float*)d_in[4];
  const float* mIm1 = (const float*)d_in[5];
  float* out = (float*)d_out;

  float* w = (float*)d_ws;
  const size_t SZ = (size_t)16 * MPAD;     // one 16 x MPAD matrix
  float* arr16 = w + 0 * SZ;
  float* Ccos  = w + 1 * SZ;
  float* Csin  = w + 2 * SZ;
  float* Tm    = w + 3 * SZ;
  float* mU    = w + 4 * SZ;
  float* Ecm   = w + 5 * SZ;
  float* Esm   = w + 6 * SZ;
  float* trigG = w + 7 * SZ;               // 2*4097 floats
  float* xgG   = trigG + 2 * (HALFM + 1);  // 8193 floats  (~3.6 MB total)

  const int NTILE_WG = ((MM + 15) / 16 + 7) / 8;   // 65 workgroups, 8 waves each

  k0_tables<<<dim3((MM + 255) / 256), 256, 0, stream>>>(trigG, xgG);
  k1_spread<<<dim3((MPAD + 255) / 256, 16), 256, 0, stream>>>(x, sig, xgG, arr16);
  trig_gemm<<<dim3(NTILE_WG), 256, 0, stream>>>(arr16, arr16, trigG, Ccos, Csin);
  k3_pointwise<<<dim3((NB * MPAD + 255) / 256), 256, 0, stream>>>(
      Ccos, Csin, mRe0, mIm0, mRe1, mIm1, Tm, mU);
  trig_gemm<<<dim3(NTILE_WG), 256, 0, stream>>>(Tm, mU, trigG, Ecm, Esm);
  k5_interp<<<dim3((NB * NPTS + 255) / 256), 256, 0, stream>>>(x, xgG, Ecm, Esm, out);
}